// QVAE_CF_26860725469352
// MI455X (gfx1250) — compile-verified
//
#include <hip/hip_runtime.h>
#include <math.h>

typedef __attribute__((ext_vector_type(2))) float v2f;
typedef __attribute__((ext_vector_type(8))) float v8f;

#define BATCH 4096
#define NNEG  200
#define LAT   64
#define NPART 4
#define NCENT 256
#define CDIM  16

// ---------------------------------------------------------------------------
// Kernel 0: cnorm[p][k] = sum_c centroids[p][k][c]^2   (4x256 floats)
// ---------------------------------------------------------------------------
__global__ __launch_bounds__(256)
void cnorm_kernel(const float* __restrict__ centroids, float* __restrict__ cnorm) {
    int i = blockIdx.x * blockDim.x + threadIdx.x;
    if (i < NPART * NCENT) {
        const float* c = centroids + (size_t)i * CDIM;
        float s = 0.f;
        #pragma unroll
        for (int j = 0; j < CDIM; ++j) s += c[j] * c[j];
        cnorm[i] = s;
    }
}

// ---------------------------------------------------------------------------
// Kernel 1: gumbel-argmax VQ encode using V_WMMA_F32_16X16X4_F32.
// One block = 16 batch rows; one wave = one partition.
// D = A(16x16 ue-slice) x B(16x16 centroid-slice^T) via 4 k-steps of 16x16x4.
// Then s = 2*dot - ||c||^2 + (-log(-log(u))), per-row argmax over 256 k,
// and scatter winning centroid into user_vec (quantized user embedding).
// ---------------------------------------------------------------------------
__global__ __launch_bounds__(128)
void encode_kernel(const int*   __restrict__ user_id,
                   const float* __restrict__ user_table,
                   const float* __restrict__ centroids,
                   const float* __restrict__ gumbel_u,
                   const float* __restrict__ cnorm,
                   float*       __restrict__ user_vec) {
    __shared__ float s_user[16][LAT];   // 4 KB: 16 gathered user rows

    const int b0  = blockIdx.x * 16;
    const int tid = threadIdx.x;

    // cooperative gather: 16 rows x 16 float4 = 256 float4, 128 threads x 2
    for (int e = tid; e < 256; e += 128) {
        int row = e >> 4, c4 = e & 15;
        int uid = user_id[b0 + row];
        float4 v = *(const float4*)(user_table + (size_t)uid * LAT + c4 * 4);
        *(float4*)(&s_user[row][c4 * 4]) = v;
    }
    __syncthreads();

    const int p    = tid >> 5;     // partition handled by this wave
    const int lane = tid & 31;
    const int h    = lane >> 4;    // half-wave id
    const int l16  = lane & 15;

    // A fragments (f32 16x4 layout): lane holds A[M=l16][k=4s+2h], [k+1]
    float ax[4], ay[4];
    #pragma unroll
    for (int s = 0; s < 4; ++s) {
        ax[s] = s_user[l16][p * CDIM + 4 * s + 2 * h];
        ay[s] = s_user[l16][p * CDIM + 4 * s + 2 * h + 1];
    }

    float best_v[8];
    int   best_i[8];
    #pragma unroll
    for (int r = 0; r < 8; ++r) { best_v[r] = -3.0e38f; best_i[r] = 0; }

    for (int nt = 0; nt < 16; ++nt) {          // 16 tiles of 16 centroids
        const int n = nt * 16 + l16;
        const float* cp = centroids + ((size_t)(p * NCENT + n)) * CDIM;

        v8f acc = {0.f, 0.f, 0.f, 0.f, 0.f, 0.f, 0.f, 0.f};
        #pragma unroll
        for (int s = 0; s < 4; ++s) {
            v2f a; a.x = ax[s];              a.y = ay[s];
            v2f b; b.x = cp[4 * s + 2 * h];  b.y = cp[4 * s + 2 * h + 1];
            acc = __builtin_amdgcn_wmma_f32_16x16x4_f32(
                      false, a, false, b, (short)0, acc, false, false);
        }

        const float cn = cnorm[p * NCENT + n];
        // C/D layout: VGPR r, lane L -> M = r + 8*(L>=16), N = L&15
        #pragma unroll
        for (int r = 0; r < 8; ++r) {
            int brow = b0 + r + 8 * h;
            float u  = gumbel_u[((size_t)p * BATCH + brow) * NCENT + n];
            float g  = -__logf(-__logf(u));
            float sc = 2.0f * acc[r] - cn + g;
            if (sc > best_v[r]) { best_v[r] = sc; best_i[r] = n; }
        }
    }

    // per-row argmax across the 16 lanes of each half (ties -> lower index)
    #pragma unroll
    for (int r = 0; r < 8; ++r) {
        float bv = best_v[r];
        int   bi = best_i[r];
        #pragma unroll
        for (int m = 1; m < 16; m <<= 1) {
            float ov = __shfl_xor(bv, m, 32);
            int   oi = __shfl_xor(bi, m, 32);
            if (ov > bv || (ov == bv && oi < bi)) { bv = ov; bi = oi; }
        }
        const int mrow = r + 8 * h;          // batch row within tile
        // lanes of a half write the winning centroid's 16 components
        user_vec[(size_t)(b0 + mrow) * LAT + p * CDIM + l16] =
            centroids[((size_t)(p * NCENT + bi)) * CDIM + l16];
    }
}

// ---------------------------------------------------------------------------
// Kernel 2: scores. One block per batch row; thread t scores item t
// (t==0 -> positive, 1..200 -> negatives). item_table is L2-resident.
// ---------------------------------------------------------------------------
__global__ __launch_bounds__(256)
void score_kernel(const int*   __restrict__ pos_id,
                  const int*   __restrict__ neg_ids,
                  const float* __restrict__ item_table,
                  const float* __restrict__ user_vec,
                  float*       __restrict__ out) {
    __shared__ float s_u[LAT];
    const int b = blockIdx.x;
    const int t = threadIdx.x;
    if (t < LAT) s_u[t] = user_vec[(size_t)b * LAT + t];
    __syncthreads();
    if (t > NNEG) return;

    const int id = (t == 0) ? pos_id[b] : neg_ids[(size_t)b * NNEG + (t - 1)];
    const float* it = item_table + (size_t)id * LAT;

    float acc = 0.f;
    #pragma unroll
    for (int j = 0; j < LAT; j += 4) {
        float4 v = *(const float4*)(it + j);
        acc += v.x * s_u[j] + v.y * s_u[j + 1] + v.z * s_u[j + 2] + v.w * s_u[j + 3];
    }

    if (t == 0) out[b];                      // (dead read avoided below)
    if (t == 0) out[b] = acc;                                   // pos_scores (B,1)
    else        out[BATCH + (size_t)b * NNEG + (t - 1)] = acc;  // neg_scores (B,200)
}

// ---------------------------------------------------------------------------
extern "C" void kernel_launch(void* const* d_in, const int* in_sizes, int n_in,
                              void* d_out, int out_size, void* d_ws, size_t ws_size,
                              hipStream_t stream) {
    const int*   user_id    = (const int*)  d_in[0];
    const int*   pos_id     = (const int*)  d_in[1];
    const int*   neg_ids    = (const int*)  d_in[2];
    const float* user_table = (const float*)d_in[3];
    const float* centroids  = (const float*)d_in[4];
    const float* item_table = (const float*)d_in[5];
    const float* gumbel_u   = (const float*)d_in[6];
    float* out = (float*)d_out;

    float* cnorm    = (float*)d_ws;                                   // 4 KB
    float* user_vec = (float*)((char*)d_ws + NPART * NCENT * sizeof(float)); // 1 MB

    cnorm_kernel <<<(NPART * NCENT + 255) / 256, 256, 0, stream>>>(centroids, cnorm);
    encode_kernel<<<BATCH / 16, 128, 0, stream>>>(user_id, user_table, centroids,
                                                  gumbel_u, cnorm, user_vec);
    score_kernel <<<BATCH, 256, 0, stream>>>(pos_id, neg_ids, item_table,
                                             user_vec, out);
}